// GeometricPositionalFingerprinter_26104811225653
// MI455X (gfx1250) — compile-verified
//
#include <hip/hip_runtime.h>

typedef __attribute__((ext_vector_type(2))) float v2f;
typedef __attribute__((ext_vector_type(8))) float v8f;

#define NV 5
#define DIM 64
#define BATCH_PER_WAVE 3
#define WAVES_PER_BLOCK 8
#define BATCH_PER_BLOCK 24   // 3 batches per wave * 8 waves
#define THREADS 256

__device__ __forceinline__ float sigmoidf(float x) {
    return 1.0f / (1.0f + __expf(-x));
}

__global__ __launch_bounds__(THREADS)
void geo_fingerprint_kernel(const float* __restrict__ verts,
                            const long long* __restrict__ ids,
                            float* __restrict__ out, int Btot) {
    // 24 batches * 5 verts * 64 dims = 7680 floats = 1920 float4 (30 KB)
    __shared__ float4 sV4[BATCH_PER_BLOCK * NV * DIM / 4];
    // 8 waves * 16x16 f32 Gram tiles (8 KB)
    __shared__ float sG[WAVES_PER_BLOCK * 16 * 16];

    const int tid = threadIdx.x;

    // ---- Stage vertices into LDS with coalesced float4 loads ----
    const float4* gV4 = (const float4*)verts;
    const int nF4Block = BATCH_PER_BLOCK * NV * DIM / 4;          // 1920
    const long long totF4 = (long long)Btot * (NV * DIM / 4);
    const long long base4 = (long long)blockIdx.x * nF4Block;
    for (int i = tid; i < nF4Block; i += THREADS) {
        long long g = base4 + i;
        float4 v = make_float4(0.f, 0.f, 0.f, 0.f);
        if (g < totF4) v = gV4[g];
        sV4[i] = v;
    }
    __syncthreads();

    // ---- Per-wave Gram matrix via V_WMMA_F32_16X16X4_F32 ----
    // Rows 0..14 of the 16x16 tile = 3 batches * 5 vertices; row 15 is junk
    // (clamped to row 14; its outputs are never read).
    // f32 16x4 A layout: lanes 0-15 -> M=0..15 (K=k0,k0+1), lanes 16-31 -> same M (K=k0+2,k0+3).
    // For the symmetric Gram (B = A^T columns = same vertices), the per-lane
    // B fragment equals the A fragment, so one load feeds both operands.
    const float* sV = (const float*)sV4;
    const int lane = tid & 31;
    const int wv   = tid >> 5;
    int r = lane & 15;
    if (r == 15) r = 14;                 // keep LDS reads in-bounds; row 15 discarded
    const int hi = lane >> 4;            // 0: K=k0,k0+1  1: K=k0+2,k0+3
    const float* vp = &sV[(wv * (BATCH_PER_WAVE * NV) + r) * DIM + hi * 2];

    v8f acc = {0.f, 0.f, 0.f, 0.f, 0.f, 0.f, 0.f, 0.f};
#pragma unroll
    for (int kc = 0; kc < DIM / 4; ++kc) {
        v2f a;
        a.x = vp[kc * 4 + 0];
        a.y = vp[kc * 4 + 1];
        acc = __builtin_amdgcn_wmma_f32_16x16x4_f32(
            /*neg_a=*/false, a, /*neg_b=*/false, a,
            /*c_mod=*/(short)0, acc, /*reuse_a=*/false, /*reuse_b=*/false);
    }

    // D layout: lane L holds column n=L&15, rows m = v + 8*(L>=16), v=0..7
#pragma unroll
    for (int v = 0; v < 8; ++v) {
        int m = v + 8 * hi;
        sG[wv * 256 + m * 16 + (lane & 15)] = acc[v];
    }
    __syncthreads();

    // ---- Scalar tail: one thread per batch (24 per block) ----
    if (tid < BATCH_PER_BLOCK) {
        long long b = (long long)blockIdx.x * BATCH_PER_BLOCK + tid;
        if (b < (long long)Btot) {
            const int w2  = tid / 3;
            const int off = (tid % 3) * NV;
            const float* Gt = &sG[w2 * 256];

            float g[5][5];
#pragma unroll
            for (int i = 0; i < 5; ++i)
#pragma unroll
                for (int j = 0; j < 5; ++j)
                    g[i][j] = Gt[(off + i) * 16 + (off + j)];

            float nsq[5];
#pragma unroll
            for (int i = 0; i < 5; ++i) nsq[i] = g[i][i];

            float d2[5][5];
#pragma unroll
            for (int i = 0; i < 5; ++i)
#pragma unroll
                for (int j = 0; j < 5; ++j)
                    d2[i][j] = (i == j) ? 0.f
                             : fmaxf(nsq[i] + nsq[j] - 2.f * g[i][j], 0.f);

            // edge statistics over 10 unique edges (unbiased std)
            float e[10];
            float esum = 0.f;
            int k = 0;
#pragma unroll
            for (int i = 0; i < 5; ++i)
#pragma unroll
                for (int j = 0; j < 5; ++j)
                    if (j > i) { e[k] = sqrtf(d2[i][j]); esum += e[k]; ++k; }
            float emean = esum * 0.1f;
            float ev = 0.f;
#pragma unroll
            for (int q = 0; q < 10; ++q) { float d = e[q] - emean; ev += d * d; }
            float estd = sqrtf(ev * (1.f / 9.f));

            // vertex spread: std of distances to centroid, all from Gram
            float rs[5], tot = 0.f;
#pragma unroll
            for (int i = 0; i < 5; ++i) {
                float s = 0.f;
#pragma unroll
                for (int j = 0; j < 5; ++j) s += g[i][j];
                rs[i] = s; tot += s;
            }
            float dd[5], dsum = 0.f;
#pragma unroll
            for (int i = 0; i < 5; ++i) {
                float dc2 = nsq[i] - 0.4f * rs[i] + tot * 0.04f;
                dd[i] = sqrtf(fmaxf(dc2, 0.f));
                dsum += dd[i];
            }
            float dmean = dsum * 0.2f;
            float dv = 0.f;
#pragma unroll
            for (int i = 0; i < 5; ++i) { float t = dd[i] - dmean; dv += t * t; }
            float spread = sqrtf(dv * 0.25f);

            // Cayley-Menger 6x6 determinant: rows 0/1 pre-swapped so the
            // first pivot is 1 (det picks up a -1); unrolled Gaussian elim.
            float Mx[6][6];
            Mx[0][0] = 1.f; Mx[1][0] = 0.f;
#pragma unroll
            for (int j = 0; j < 5; ++j) { Mx[0][j + 1] = d2[0][j]; Mx[1][j + 1] = 1.f; }
#pragma unroll
            for (int i = 1; i < 5; ++i) {
                Mx[i + 1][0] = 1.f;
#pragma unroll
                for (int j = 0; j < 5; ++j) Mx[i + 1][j + 1] = d2[i][j];
            }
            float det = -1.f;
#pragma unroll
            for (int kk = 0; kk < 6; ++kk) {
                float pk = Mx[kk][kk];
                det *= pk;
                float inv = 1.f / pk;
#pragma unroll
                for (int i = 0; i < 6; ++i) {
                    if (i > kk) {
                        float f = Mx[i][kk] * inv;
#pragma unroll
                        for (int j = 0; j < 6; ++j)
                            if (j > kk) Mx[i][j] -= f * Mx[kk][j];
                    }
                }
            }
            float volume = sqrtf(fmaxf(det * (-1.f / 9216.f), 0.f));

            float vn = sigmoidf(volume * 10.0f);
            float er = sigmoidf(estd / (emean + 1e-6f));
            float sn = sigmoidf(spread);
            float seed = 0.4f * vn + 0.3f * er + 0.3f * sn;

            long long aid = ids[b];
            float idc = (float)((aid * 2654435761LL) % 1000000LL) * 1e-6f;
            seed = 0.1f * seed + 0.9f * idc;
            seed = fminf(fmaxf(seed, 1e-6f), 1.f - 1e-6f);

            // Cantor digit extraction, depth 8
            float x = seed, cv = 0.f, factor = 0.5f;
#pragma unroll
            for (int it = 0; it < 8; ++it) {
                float xs = x * 3.f;
                int dg = (int)xs;          // trunc == floor for x >= 0
                x = xs - (float)dg;
                if (dg == 2) cv += factor;
                factor *= 0.5f;
            }
            out[b] = fminf(fmaxf(cv, 0.f), 1.f);
        }
    }
}

extern "C" void kernel_launch(void* const* d_in, const int* in_sizes, int n_in,
                              void* d_out, int out_size, void* d_ws, size_t ws_size,
                              hipStream_t stream) {
    const float*     verts = (const float*)d_in[0];      // [B, 5, 64] f32
    const long long* ids   = (const long long*)d_in[1];  // [B] int64
    float*           out   = (float*)d_out;              // [B] f32
    const int Btot = in_sizes[1];
    const int nBlocks = (Btot + BATCH_PER_BLOCK - 1) / BATCH_PER_BLOCK;
    geo_fingerprint_kernel<<<nBlocks, THREADS, 0, stream>>>(verts, ids, out, Btot);
}